// SparsifiedLinear_79508434583776
// MI455X (gfx1250) — compile-verified
//
#include <hip/hip_runtime.h>
#include <stdint.h>

#define MDIM  8192
#define KDIM  8192
#define NNZ   256
#define BATCH 32
#define WPB   8        // waves per block
#define CHUNK 32       // nonzeros staged per async chunk (== wave width)
#define NCHUNK (NNZ / CHUNK)

// One wave per sparse row; lane = batch element.
// (idx,val) metadata double-buffered into LDS with CDNA5 async-to-LDS loads.
template <bool TRANSPOSE_OUT>
__global__ __launch_bounds__(256, 4) void ell_spmv_kernel(
    const int* __restrict__ idxmat,    // [rows, NNZ]
    const float* __restrict__ valmat,  // [rows, NNZ]
    const float* __restrict__ src,     // [*, BATCH] gather source
    float* __restrict__ dst)           // [rows, BATCH] or [BATCH, MDIM]
{
    __shared__ int2 s_pair[WPB][2][CHUNK];   // {idx, val-bits} per staged nonzero

    const int lane = threadIdx.x & 31;
    const int wave = threadIdx.x >> 5;
    const int row  = blockIdx.x * WPB + wave;

    // Per-lane global byte offset of element (row*NNZ + c*CHUNK + lane)
    const int base_byte = (row * NNZ + lane) * 4;

    // LDS byte offsets of this wave's two staging buffers (lane's slot)
    const uint32_t lds0 = (uint32_t)(uintptr_t)&s_pair[wave][0][lane].x;
    const uint32_t lds1 = (uint32_t)(uintptr_t)&s_pair[wave][1][lane].x;

    auto issue_chunk = [&](int c) {
        const uint32_t lds  = (c & 1) ? lds1 : lds0;
        const int      goff = base_byte + c * CHUNK * 4;
        // idx dword -> s_pair[wave][buf][lane].x
        asm volatile("global_load_async_to_lds_b32 %0, %1, %2"
                     :: "v"(lds), "v"(goff), "s"(idxmat) : "memory");
        // val dword -> s_pair[wave][buf][lane].y
        asm volatile("global_load_async_to_lds_b32 %0, %1, %2"
                     :: "v"(lds + 4u), "v"(goff), "s"(valmat) : "memory");
    };

    float acc0 = 0.f, acc1 = 0.f, acc2 = 0.f, acc3 = 0.f;

    issue_chunk(0);
    for (int c = 0; c < NCHUNK; ++c) {
        if (c + 1 < NCHUNK) {
            issue_chunk(c + 1);                      // up to 4 async ops in flight
            asm volatile("s_wait_asynccnt 0x2" ::: "memory");  // chunk c landed
        } else {
            asm volatile("s_wait_asynccnt 0x0" ::: "memory");
        }
        const int buf = c & 1;
#pragma unroll
        for (int j = 0; j < CHUNK; j += 4) {
            int2 p0 = s_pair[wave][buf][j + 0];
            int2 p1 = s_pair[wave][buf][j + 1];
            int2 p2 = s_pair[wave][buf][j + 2];
            int2 p3 = s_pair[wave][buf][j + 3];
            // LDS value is wave-uniform; readfirstlane moves it to SGPRs so the
            // gather base is SALU math + saddr-form loads, val is SGPR src to fmac.
            int   i0 = __builtin_amdgcn_readfirstlane(p0.x);
            float v0 = __int_as_float(__builtin_amdgcn_readfirstlane(p0.y));
            int   i1 = __builtin_amdgcn_readfirstlane(p1.x);
            float v1 = __int_as_float(__builtin_amdgcn_readfirstlane(p1.y));
            int   i2 = __builtin_amdgcn_readfirstlane(p2.x);
            float v2 = __int_as_float(__builtin_amdgcn_readfirstlane(p2.y));
            int   i3 = __builtin_amdgcn_readfirstlane(p3.x);
            float v3 = __int_as_float(__builtin_amdgcn_readfirstlane(p3.y));
            acc0 = fmaf(v0, src[(size_t)i0 * BATCH + lane], acc0);
            acc1 = fmaf(v1, src[(size_t)i1 * BATCH + lane], acc1);
            acc2 = fmaf(v2, src[(size_t)i2 * BATCH + lane], acc2);
            acc3 = fmaf(v3, src[(size_t)i3 * BATCH + lane], acc3);
        }
    }
    const float acc = (acc0 + acc1) + (acc2 + acc3);

    if constexpr (!TRANSPOSE_OUT) {
        // bx[row, batch] — coalesced 128B per wave
        dst[(size_t)row * BATCH + lane] = acc;
    } else {
        // transpose through padded LDS tile -> out[batch, m] coalesced-ish
        __shared__ float s_tile[BATCH][WPB + 1];
        s_tile[lane][wave] = acc;
        __syncthreads();
        const int b  = threadIdx.x >> 3;  // 0..31
        const int ml = threadIdx.x & 7;   // 0..7
        dst[(size_t)b * MDIM + (size_t)blockIdx.x * WPB + ml] = s_tile[b][ml];
    }
}

extern "C" void kernel_launch(void* const* d_in, const int* in_sizes, int n_in,
                              void* d_out, int out_size, void* d_ws, size_t ws_size,
                              hipStream_t stream) {
    (void)in_sizes; (void)n_in; (void)out_size; (void)ws_size;
    const float* x      = (const float*)d_in[0];
    const int*   a_idx  = (const int*)  d_in[1];
    const float* a_vals = (const float*)d_in[2];
    const int*   b_idx  = (const int*)  d_in[3];
    const float* b_vals = (const float*)d_in[4];
    float* out = (float*)d_out;
    float* bx  = (float*)d_ws;   // KDIM * BATCH fp32 = 1 MB intermediate

    dim3 block(32 * WPB);
    // Stage 1: bx = B @ x
    ell_spmv_kernel<false><<<dim3(KDIM / WPB), block, 0, stream>>>(b_idx, b_vals, x, bx);
    // Stage 2: out[b, m] = (A @ bx)[m, b]
    ell_spmv_kernel<true ><<<dim3(MDIM / WPB), block, 0, stream>>>(a_idx, a_vals, bx, out);
}